// ToBEVReduction_43817256354331
// MI455X (gfx1250) — compile-verified
//
#include <hip/hip_runtime.h>
#include <hip/hip_bf16.h>
#include <stdint.h>

// ---------------------------------------------------------------------------
// ToBEVReduction for MI455X (gfx1250, wave32).
// Bitmap-rank replaces sort:  rank(key) = #set bits below `key` in a 2^27-bit
// presence bitmap == index into sorted-unique keys (matches jnp.unique order).
// ---------------------------------------------------------------------------

#define GRID_DIM   512
#define KEYSPACE   (512u * 512u * 512u)     // 2^27
#define NWORDS     (KEYSPACE / 32u)          // 4,194,304 words (16 MB)
#define SCAN_WPB   4096                      // words per scan block
#define SCAN_BLKS  (NWORDS / SCAN_WPB)       // 1024 scan blocks

typedef float    nfloat4 __attribute__((ext_vector_type(4)));  // native vecs for NT ops
typedef uint32_t nuint4  __attribute__((ext_vector_type(4)));

// -------------------------- init: zero everything --------------------------
// b128 non-temporal streaming stores: 205 MB zero of the accumulation target
// must not evict the L2-resident bitmap/prefix arrays.
__global__ void k_init(float* __restrict__ outF, uint32_t* __restrict__ bitmap,
                       uint32_t* __restrict__ counts, uint32_t* __restrict__ uniq,
                       size_t nF4, int nPts) {
    size_t idx    = (size_t)blockIdx.x * blockDim.x + threadIdx.x;
    size_t stride = (size_t)gridDim.x * blockDim.x;
    nfloat4* outV = (nfloat4*)outF;
    nuint4*  bmV  = (nuint4*)bitmap;
    const nfloat4 zf = {0.f, 0.f, 0.f, 0.f};
    const nuint4  zu = {0u, 0u, 0u, 0u};
    for (size_t k = idx; k < nF4; k += stride)
        __builtin_nontemporal_store(zf, &outV[k]);            // global_store_b128 NT
    for (size_t k = idx; k < (size_t)(NWORDS / 4); k += stride)
        bmV[k] = zu;                                          // keep RT: L2 resident
    for (size_t k = idx; k < (size_t)nPts; k += stride) {
        counts[k] = 0u;
        uniq[k]   = 0xFFFFFFFFu;                              // sentinel key -1
    }
}

// ---------------- pass 1: build keys, set presence bitmap ------------------
__global__ void k_keys(const int* __restrict__ coords, uint32_t* __restrict__ keys,
                       uint32_t* __restrict__ bitmap, int nPts) {
    int i = blockIdx.x * blockDim.x + threadIdx.x;
    if (i >= nPts) return;
    int4 c = ((const int4*)coords)[i];
    // non_z dims = columns {0,2,3};  key = (c0*512 + c1)*512 + c2
    uint32_t key = ((((uint32_t)c.x << 9) | (uint32_t)c.z) << 9) | (uint32_t)c.w;
    keys[i] = key;
    atomicOr(&bitmap[key >> 5], 1u << (key & 31u));
}

// ---------------- pass 2a: per-block popcount sums -------------------------
__global__ void k_scanA(const uint32_t* __restrict__ bitmap,
                        uint32_t* __restrict__ blocksums) {
    __shared__ uint32_t sh[256];
    uint32_t base = blockIdx.x * SCAN_WPB + threadIdx.x * 16;
    const uint4* p = (const uint4*)(bitmap + base);
    uint32_t s = 0;
#pragma unroll
    for (int j = 0; j < 4; ++j) {
        uint4 w = p[j];
        s += __popc(w.x) + __popc(w.y) + __popc(w.z) + __popc(w.w);
    }
    sh[threadIdx.x] = s;
    __syncthreads();
    for (int off = 128; off > 0; off >>= 1) {
        if ((int)threadIdx.x < off) sh[threadIdx.x] += sh[threadIdx.x + off];
        __syncthreads();
    }
    if (threadIdx.x == 0) blocksums[blockIdx.x] = sh[0];
}

// ---------------- pass 2b: exclusive scan of 1024 block sums ---------------
__global__ void k_scanB(uint32_t* __restrict__ blocksums) {
    __shared__ uint32_t sh[SCAN_BLKS];
    uint32_t v = blocksums[threadIdx.x];
    sh[threadIdx.x] = v;
    __syncthreads();
    for (int off = 1; off < SCAN_BLKS; off <<= 1) {
        uint32_t t = (threadIdx.x >= (uint32_t)off) ? sh[threadIdx.x - off] : 0u;
        __syncthreads();
        sh[threadIdx.x] += t;
        __syncthreads();
    }
    blocksums[threadIdx.x] = sh[threadIdx.x] - v;             // inclusive -> exclusive
}

// ---------------- pass 2c: per-word exclusive prefix -----------------------
__global__ void k_scanC(const uint32_t* __restrict__ bitmap,
                        const uint32_t* __restrict__ blocksums,
                        uint32_t* __restrict__ wordpfx) {
    __shared__ uint32_t sh[256];
    uint32_t base = blockIdx.x * SCAN_WPB + threadIdx.x * 16;
    const uint4* p = (const uint4*)(bitmap + base);
    uint4 w[4];
    uint32_t s = 0;
#pragma unroll
    for (int j = 0; j < 4; ++j) {
        w[j] = p[j];
        s += __popc(w[j].x) + __popc(w[j].y) + __popc(w[j].z) + __popc(w[j].w);
    }
    sh[threadIdx.x] = s;
    __syncthreads();
    for (int off = 1; off < 256; off <<= 1) {                 // Hillis-Steele inclusive
        uint32_t t = (threadIdx.x >= (uint32_t)off) ? sh[threadIdx.x - off] : 0u;
        __syncthreads();
        sh[threadIdx.x] += t;
        __syncthreads();
    }
    uint32_t run = blocksums[blockIdx.x] + sh[threadIdx.x] - s;  // block base + local excl
#pragma unroll
    for (int j = 0; j < 4; ++j) {
        uint32_t c[4] = { w[j].x, w[j].y, w[j].z, w[j].w };
#pragma unroll
        for (int q = 0; q < 4; ++q) {
            wordpfx[base + j * 4 + q] = run;
            run += __popc(c[q]);
        }
    }
}

// ------- pass 3: rank lookup + async row fetch to LDS + atomic scatter -----
// One wave (32 lanes) per point: 32 lanes x b128 = 512 B = one feature row.
__global__ void k_accum(const float* __restrict__ features,
                        const uint32_t* __restrict__ keys,
                        const uint32_t* __restrict__ bitmap,
                        const uint32_t* __restrict__ wordpfx,
                        uint32_t* __restrict__ counts,
                        uint32_t* __restrict__ uniq,
                        float* __restrict__ outF, int nPts) {
    __shared__ float4 fbuf[8][32];                            // 8 waves * 512 B
    const int wave = threadIdx.x >> 5;
    const int lane = threadIdx.x & 31;
    const int i = blockIdx.x * 8 + wave;
    if (i >= nPts) return;

    // CDNA5 async path: start the 16 B/lane feature fetch into LDS, then
    // overlap the two dependent rank-lookup loads with it.
    uint32_t lds_addr = (uint32_t)(uintptr_t)&fbuf[wave][lane];
    unsigned long long gaddr =
        (unsigned long long)(((const float4*)features) + (size_t)i * 32 + lane);
    asm volatile("global_load_async_to_lds_b128 %0, %1, off"
                 :: "v"(lds_addr), "v"(gaddr) : "memory");

    uint32_t key  = keys[i];
    uint32_t word = bitmap[key >> 5];
    uint32_t pre  = wordpfx[key >> 5];
    uint32_t rank = pre + __popc(word & ((1u << (key & 31u)) - 1u));

    if (lane == 0) {
        atomicAdd(&counts[rank], 1u);
        uniq[rank] = key;                                     // same value from all writers
    }

    asm volatile("s_wait_asynccnt 0x0" ::: "memory");
    float4 f = fbuf[wave][lane];

    float* dst = outF + (size_t)rank * 128 + lane * 4;
    unsafeAtomicAdd(dst + 0, f.x);                            // global_atomic_add_f32 (no-return)
    unsafeAtomicAdd(dst + 1, f.y);
    unsafeAtomicAdd(dst + 2, f.z);
    unsafeAtomicAdd(dst + 3, f.w);
}

// -------- pass 4: divide by count, decode coords, emit cnt -----------------
__global__ void k_final(float* __restrict__ out,
                        const uint32_t* __restrict__ counts,
                        const uint32_t* __restrict__ uniq, int nPts) {
    const int wave = threadIdx.x >> 5;
    const int lane = threadIdx.x & 31;
    const int r = blockIdx.x * 8 + wave;
    if (r >= nPts) return;

    float cnt = (float)counts[r];
    float inv = 1.0f / fmaxf(cnt, 1.0f);

    nfloat4* rowp = ((nfloat4*)out) + (size_t)r * 32 + lane;
    nfloat4 v = *rowp;
    v *= inv;
    __builtin_nontemporal_store(v, rowp);

    if (lane == 0) {
        int k = (int)uniq[r];
        int d0, d1, d2;
        if (k < 0) { d0 = -1; d1 = 511; d2 = 511; }           // jnp floor-div/mod of -1
        else { d2 = k & 511; int rem = k >> 9; d1 = rem & 511; d0 = rem >> 9; }
        float* cp = out + (size_t)nPts * 128 + (size_t)r * 4;
        cp[0] = (float)d0;                                    // col0 = d0
        cp[1] = 0.0f;                                         // Z_DIM = 1 -> 0
        cp[2] = (float)d1;                                    // col2 = d1
        cp[3] = (float)d2;                                    // col3 = d2
        out[(size_t)nPts * 132 + r] = cnt;
    }
}

// ---------------------------------------------------------------------------
extern "C" void kernel_launch(void* const* d_in, const int* in_sizes, int n_in,
                              void* d_out, int out_size, void* d_ws, size_t ws_size,
                              hipStream_t stream) {
    const float* features = (const float*)d_in[0];
    const int*   coords   = (const int*)d_in[1];
    float*       out      = (float*)d_out;

    const int N = in_sizes[0] / 128;                          // 400000

    // workspace layout (uint32 units)
    uint32_t* ws        = (uint32_t*)d_ws;
    uint32_t* bitmap    = ws;                                 // 4,194,304
    uint32_t* wordpfx   = ws + NWORDS;                        // 4,194,304
    uint32_t* blocksums = ws + 2 * NWORDS;                    // 1,024
    uint32_t* counts    = ws + 2 * NWORDS + SCAN_BLKS;        // N
    uint32_t* uniq      = counts + N;                         // N
    uint32_t* keys      = uniq + N;                           // N

    const size_t nF4 = (size_t)N * 32;                        // float4 count

    k_init <<<4096, 256, 0, stream>>>(out, bitmap, counts, uniq, nF4, N);
    k_keys <<<(N + 255) / 256, 256, 0, stream>>>(coords, keys, bitmap, N);
    k_scanA<<<SCAN_BLKS, 256, 0, stream>>>(bitmap, blocksums);
    k_scanB<<<1, SCAN_BLKS, 0, stream>>>(blocksums);
    k_scanC<<<SCAN_BLKS, 256, 0, stream>>>(bitmap, blocksums, wordpfx);
    k_accum<<<(N + 7) / 8, 256, 0, stream>>>(features, keys, bitmap, wordpfx,
                                             counts, uniq, out, N);
    k_final<<<(N + 7) / 8, 256, 0, stream>>>(out, counts, uniq, N);
}